// Superfeatures_79903571575311
// MI455X (gfx1250) — compile-verified
//
#include <hip/hip_runtime.h>
#include <math.h>

// Problem constants (from the reference): feats [B=4, C=64, H=512, W=512] f32,
// labels [B, 1, H, W] int32 in [0,256). Crop 1-px border -> 510x510 pixels.
// Output [B, C, 256] f32 segment-max, empty segments = 0.

#define NSEG   256
#define CTOT   64
#define CBLK   32      // channels per workgroup
#define RBLK   8       // rows per workgroup tile
#define HDIM   512
#define WDIM   512
#define CROP   510
#define NTHR   256     // 8 wave32

typedef float v4f __attribute__((ext_vector_type(4)));
typedef int   v4i __attribute__((ext_vector_type(4)));

__global__ void sf_init_kernel(float* __restrict__ out, int n) {
    int i = blockIdx.x * blockDim.x + threadIdx.x;
    if (i < n) out[i] = -INFINITY;
}

// Test-before-atomic max: most elements lose against the running max, so a
// cheap ds_load + compare filters out the RMW; the atomic re-checks, so a
// stale read is harmless. fmax is commutative/associative -> deterministic.
__device__ __forceinline__ void seg_upd(float* __restrict__ accc, int s, float v) {
    if (v > accc[s]) {
        __hip_atomic_fetch_max(&accc[s], v,
                               __ATOMIC_RELAXED, __HIP_MEMORY_SCOPE_WORKGROUP);
    }
}

__global__ __launch_bounds__(NTHR)
void sf_segmax_kernel(const float* __restrict__ feats,
                      const int*   __restrict__ labels,
                      float*       __restrict__ out) {
    __shared__ float acc[CBLK * NSEG];   // 32 KB: per-WG [channel][segment] max
    __shared__ int   lab[RBLK * WDIM];   // 16 KB: full 512-wide label rows

    const int b   = blockIdx.z;
    const int c0  = blockIdx.y * CBLK;
    const int r0  = blockIdx.x * RBLK;                 // cropped-row index
    const int h0  = r0 + 1;                            // absolute row
    const int nrows = (CROP - r0) < RBLK ? (CROP - r0) : RBLK;
    const int tid = threadIdx.x;

    // 1) init LDS accumulator to -inf
    for (int i = tid; i < CBLK * NSEG; i += NTHR) acc[i] = -INFINITY;

    // 2) linear async global->LDS copy of the label tile (full rows, stride
    //    512 matches both global and LDS layout -> one flat loop)
    {
        const int* lbase = labels + ((size_t)b * HDIM + h0) * WDIM;
        const int  nel   = nrows * WDIM;
        for (int e = tid; e < nel; e += NTHR) {
#if __has_builtin(__builtin_amdgcn_global_load_async_to_lds_b32)
            int* gnc = const_cast<int*>(lbase + e);
            __builtin_amdgcn_global_load_async_to_lds_b32(
                (__attribute__((address_space(1))) int*)gnc,
                (__attribute__((address_space(3))) int*)(lab + e),
                /*imm_offset=*/0, /*cpol=*/0);
#else
            lab[e] = lbase[e];
#endif
        }
    }
#if __has_builtin(__builtin_amdgcn_global_load_async_to_lds_b32)
#if __has_builtin(__builtin_amdgcn_s_wait_asynccnt)
    __builtin_amdgcn_s_wait_asynccnt(0);
#else
    asm volatile("s_wait_asynccnt 0x0" ::: "memory");
#endif
#endif
    __syncthreads();

    // 3) channel-major accumulation, 16B vector loads of full rows.
    //    col = (chunk & 127)*4 + j; valid cols are [1, 510]:
    //    j==0 invalid only at chunk 0, j==3 invalid only at chunk 127.
    const int nchunk = nrows * (WDIM / 4);
    const v4i* __restrict__ l4 = (const v4i*)lab;
    for (int c = 0; c < CBLK; ++c) {
        const v4f* __restrict__ f4 =
            (const v4f*)(feats + ((size_t)(b * CTOT + c0 + c) * HDIM + h0) * WDIM);
        float* accc = acc + c * NSEG;
        for (int e = tid; e < nchunk; e += NTHR) {
            v4f v = __builtin_nontemporal_load(&f4[e]);   // streamed once: NT hint
            v4i s = l4[e];
            const int chunk = e & 127;
            if (chunk != 0)   seg_upd(accc, s.x, v.x);    // skip col 0
            seg_upd(accc, s.y, v.y);
            seg_upd(accc, s.z, v.z);
            if (chunk != 127) seg_upd(accc, s.w, v.w);    // skip col 511
        }
    }
    __syncthreads();

    // 4) combine privatized results into global output (fmax order-independent)
    float* outb = out + ((size_t)b * CTOT + c0) * NSEG;
    for (int i = tid; i < CBLK * NSEG; i += NTHR) {
        float v = acc[i];
        if (v > -INFINITY) {
            __hip_atomic_fetch_max(&outb[i], v,
                                   __ATOMIC_RELAXED, __HIP_MEMORY_SCOPE_AGENT);
        }
    }
}

__global__ void sf_finalize_kernel(float* __restrict__ out, int n) {
    int i = blockIdx.x * blockDim.x + threadIdx.x;
    if (i < n) {
        float v = out[i];
        out[i] = (v == -INFINITY) ? 0.0f : v;   // empty segments -> 0
    }
}

extern "C" void kernel_launch(void* const* d_in, const int* in_sizes, int n_in,
                              void* d_out, int out_size, void* d_ws, size_t ws_size,
                              hipStream_t stream) {
    (void)n_in; (void)d_ws; (void)ws_size;
    const float* feats  = (const float*)d_in[0];
    const int*   labels = (const int*)d_in[1];
    float*       out    = (float*)d_out;

    const int B = in_sizes[0] / (CTOT * HDIM * WDIM);   // = 4
    const int n = out_size;                             // B * CTOT * NSEG = 65536

    // init output to -inf
    sf_init_kernel<<<(n + NTHR - 1) / NTHR, NTHR, 0, stream>>>(out, n);

    // main segment-max
    dim3 grid((CROP + RBLK - 1) / RBLK,   // 64 row tiles
              CTOT / CBLK,                // 2 channel blocks
              B);                         // 4 batches
    sf_segmax_kernel<<<grid, NTHR, 0, stream>>>(feats, labels, out);

    // map -inf (empty segment) -> 0
    sf_finalize_kernel<<<(n + NTHR - 1) / NTHR, NTHR, 0, stream>>>(out, n);
}